// MaxSimParallel_30812095381663
// MI455X (gfx1250) — compile-verified
//
#include <hip/hip_runtime.h>

typedef __attribute__((ext_vector_type(2))) float v2f;
typedef __attribute__((ext_vector_type(4))) float v4f;
typedef __attribute__((ext_vector_type(8))) float v8f;

#define B_SZ 16
#define QL 32
#define H 128
#define DL 64
#define NCAND 1024
#define KTOP 100
#define CAND_PER_BLK 8

// ---------------------------------------------------------------------------
// Kernel A: pids = emb2pid[topk_indices]; validity clamp; per-row bitonic
// sort descending; duplicates -> -1.  One block per batch row.
// ---------------------------------------------------------------------------
__global__ __launch_bounds__(512) void pid_dedup_kernel(
    const int* __restrict__ topk, const int* __restrict__ emb2pid,
    int n_docs, int n_emb, int* __restrict__ pids_out) {
  __shared__ int s[NCAND];
  const int b = blockIdx.x;
  const int tid = threadIdx.x;
  for (int i = tid; i < NCAND; i += 512) {
    int e = topk[b * NCAND + i];
    int p = (e >= 0 && e < n_emb) ? emb2pid[e] : -1;
    s[i] = (p < 0 || p >= n_docs) ? -1 : p;
  }
  __syncthreads();
  for (int k = 2; k <= NCAND; k <<= 1) {
    for (int j = k >> 1; j > 0; j >>= 1) {
      for (int i = tid; i < NCAND; i += 512) {
        int ixj = i ^ j;
        if (ixj > i) {
          bool desc = ((i & k) == 0);
          int a = s[i], c = s[ixj];
          if (desc ? (a < c) : (a > c)) { s[i] = c; s[ixj] = a; }
        }
      }
      __syncthreads();
    }
  }
  for (int i = tid; i < NCAND; i += 512) {
    int v = s[i];
    if (i > 0 && s[i] == s[i - 1]) v = -1;
    pids_out[b * NCAND + i] = v;
  }
}

// ---------------------------------------------------------------------------
// Kernel B: MaxSim scores via V_WMMA_F32_16X16X4_F32.
// Block = 256 threads (8 waves) handles 8 candidates of one batch row.
// sim[32x64] split into 2x4 tiles of 16x16, one per wave; K=128 in steps of 4.
// Doc tiles streamed global->LDS with GLOBAL_LOAD_ASYNC_TO_LDS_B128
// (ASYNCcnt-tracked, no VGPR staging).
// ---------------------------------------------------------------------------
__global__ __launch_bounds__(256) void maxsim_kernel(
    const float* __restrict__ q_vectors, const float* __restrict__ vectors,
    const int* __restrict__ pids, float* __restrict__ scores) {
  __shared__ __attribute__((aligned(16))) float lQ[QL * H];   // 16 KB
  __shared__ __attribute__((aligned(16))) float lD[DL * H];   // 32 KB
  __shared__ float lSim[QL * DL];                             //  8 KB

  const int b = blockIdx.x;
  const int chunk = blockIdx.y;
  const int tid = threadIdx.x;
  const int lane = tid & 31;
  const int wave = tid >> 5;
  const int m0 = (wave & 1) * 16;   // query-token tile
  const int n0 = (wave >> 1) * 16;  // doc-token tile

  // Stage Q[b] once per block: 4096 floats as float4.
  {
    const v4f* src = (const v4f*)(q_vectors + (size_t)b * QL * H);
    v4f* dst = (v4f*)lQ;
    for (int i = tid; i < QL * H / 4; i += 256) dst[i] = src[i];
  }

  // LDS byte offset of lD: LDS aperture lives in addr[63:32], so the low
  // 32 bits of the generic pointer are the in-LDS address.
  const unsigned ldsD = (unsigned)(size_t)(void*)lD;

  // ISA A/B lane layout for 32-bit 16x4 / 4x16 operands:
  // lane%16 = matrix row (A: M, B: N), lane/16 selects K-pair {0,1} vs {2,3}.
  const int op_row = lane & 15;
  const int op_koff = (lane >> 4) * 2;

  for (int c = 0; c < CAND_PER_BLK; ++c) {
    const int cand = chunk * CAND_PER_BLK + c;
    const int pid = pids[b * NCAND + cand];  // uniform across block
    if (pid < 0) {
      if (tid == 0) scores[b * NCAND + cand] = -__builtin_inff();
      continue;
    }
    __syncthreads();  // previous iteration done reading lD
    // Stage doc tile: 32 KB = 2048 x 16B chunks, 8 async b128 ops/thread.
    {
      const char* src = (const char*)(vectors + (size_t)pid * DL * H);
#pragma unroll
      for (int it = 0; it < 8; ++it) {
        const int byteOff = (tid + it * 256) * 16;
        unsigned long long ga = (unsigned long long)(size_t)(src + byteOff);
        unsigned la = ldsD + (unsigned)byteOff;
        asm volatile("global_load_async_to_lds_b128 %0, %1, off"
                     :: "v"(la), "v"(ga) : "memory");
      }
    }
    // Prefetch next candidate's doc tile into GL2 (32 KB = 256 x 128B lines).
    if (c + 1 < CAND_PER_BLK) {
      int pn = pids[b * NCAND + cand + 1];
      if (pn >= 0) {
        const char* p = (const char*)(vectors + (size_t)pn * DL * H);
        __builtin_prefetch(p + tid * 128, 0, 0);
      }
    }
    asm volatile("s_wait_asynccnt 0x0" ::: "memory");
    __syncthreads();

    v8f acc = {};
    const v2f* qa = (const v2f*)(lQ + (m0 + op_row) * H + op_koff);
    const v2f* db = (const v2f*)(lD + (n0 + op_row) * H + op_koff);
#pragma unroll
    for (int k0 = 0; k0 < H; k0 += 4) {
      v2f a = qa[k0 >> 1];
      v2f bb = db[k0 >> 1];
      acc = __builtin_amdgcn_wmma_f32_16x16x4_f32(
          /*neg_a=*/false, a, /*neg_b=*/false, bb,
          /*c_mod=*/(short)0, acc, /*reuse_a=*/false, /*reuse_b=*/false);
    }

    // C/D layout: VGPR r, lanes 0-15 -> M=r, lanes 16-31 -> M=8+r; N=lane%16.
    {
      const int col = n0 + op_row;
      const int rbase = m0 + ((lane < 16) ? 0 : 8);
#pragma unroll
      for (int r = 0; r < 8; ++r) lSim[(rbase + r) * DL + col] = acc[r];
    }
    __syncthreads();

    // max over doc tokens, mean over query tokens (wave 0 only).
    if (wave == 0) {
      float m = -__builtin_inff();
      const float* row = lSim + lane * DL;
#pragma unroll 8
      for (int d = 0; d < DL; ++d) m = fmaxf(m, row[d]);
      for (int off = 16; off > 0; off >>= 1) m += __shfl_xor(m, off, 32);
      if (lane == 0) scores[b * NCAND + cand] = m * (1.0f / QL);
    }
    // next iteration's leading __syncthreads keeps lSim/lD safe
  }
}

// ---------------------------------------------------------------------------
// Kernel C: per-row top-100 via bitonic sort of (score desc, idx asc).
// ---------------------------------------------------------------------------
__global__ __launch_bounds__(512) void topk_kernel(
    const float* __restrict__ scores, const int* __restrict__ pids,
    float* __restrict__ out) {
  __shared__ float sc[NCAND];
  __shared__ int si[NCAND];
  const int b = blockIdx.x;
  const int tid = threadIdx.x;
  for (int i = tid; i < NCAND; i += 512) {
    sc[i] = scores[b * NCAND + i];
    si[i] = i;
  }
  __syncthreads();
  for (int k = 2; k <= NCAND; k <<= 1) {
    for (int j = k >> 1; j > 0; j >>= 1) {
      for (int i = tid; i < NCAND; i += 512) {
        int ixj = i ^ j;
        if (ixj > i) {
          bool desc = ((i & k) == 0);
          float s1 = sc[i], s2 = sc[ixj];
          int i1 = si[i], i2 = si[ixj];
          bool better = (s1 > s2) || (s1 == s2 && i1 < i2);
          if (desc ? !better : better) {
            sc[i] = s2; sc[ixj] = s1;
            si[i] = i2; si[ixj] = i1;
          }
        }
      }
      __syncthreads();
    }
  }
  if (tid < KTOP) {
    int idx = si[tid];
    out[b * KTOP + tid] = (float)pids[b * NCAND + idx];        // top_pids
    out[B_SZ * KTOP + b * KTOP + tid] = sc[tid];               // top_scores
  }
}

// ---------------------------------------------------------------------------
extern "C" void kernel_launch(void* const* d_in, const int* in_sizes, int n_in,
                              void* d_out, int out_size, void* d_ws,
                              size_t ws_size, hipStream_t stream) {
  const float* q_vectors = (const float*)d_in[0];
  const int* topk = (const int*)d_in[1];
  const float* vectors = (const float*)d_in[2];
  const int* emb2pid = (const int*)d_in[3];
  // d_in[4] = k (scalar); reference takes min(k, 1024) = 100, baked as KTOP.

  const int n_docs = in_sizes[2] / (DL * H);  // 20000
  const int n_emb = in_sizes[3];              // 1,280,000

  int* pids_ws = (int*)d_ws;                                       // 64 KB
  float* scores_ws = (float*)((char*)d_ws + B_SZ * NCAND * sizeof(int));

  pid_dedup_kernel<<<dim3(B_SZ), dim3(512), 0, stream>>>(
      topk, emb2pid, n_docs, n_emb, pids_ws);
  maxsim_kernel<<<dim3(B_SZ, NCAND / CAND_PER_BLK), dim3(256), 0, stream>>>(
      q_vectors, vectors, pids_ws, scores_ws);
  topk_kernel<<<dim3(B_SZ), dim3(512), 0, stream>>>(
      scores_ws, pids_ws, (float*)d_out);
}